// NSAMSA_55611236549028
// MI455X (gfx1250) — compile-verified
//
#include <hip/hip_runtime.h>
#include <math.h>

typedef float v2f __attribute__((ext_vector_type(2)));
typedef float v8f __attribute__((ext_vector_type(8)));

#define DIMX   512
#define NH     8
#define HE     64
#define MSEL   16
#define NPTS   4096
#define NBALL  256
#define NB     2
#define SZROW  ((size_t)NB * NPTS)          // 8192 rows
#define SZMAT  (SZROW * DIMX)               // 8192*512

// ---- WMMA fragment loaders (V_WMMA_F32_16X16X4_F32 layouts, wave32) ----
// A (16x4, MxK): lane l -> m=l&15, VGPR0/1 = K = 2*(l>>4) + {0,1}
__device__ __forceinline__ v2f frag_a(const float* __restrict__ A, int lda,
                                      int row0, int kk, int lane) {
    int m = lane & 15, kh = lane >> 4;
    const float* p = A + (size_t)(row0 + m) * lda + kk + 2 * kh;
    v2f a; a.x = p[0]; a.y = p[1]; return a;
}
// B (4x16, KxN): lane l -> n=l&15, VGPR0/1 = K = 2*(l>>4) + {0,1}
__device__ __forceinline__ v2f frag_b(const float* __restrict__ Bm, int ldb,
                                      int col0, int kk, int lane) {
    int nn = lane & 15, kh = lane >> 4;
    const float* p = Bm + (size_t)(kk + 2 * kh) * ldb + col0 + nn;
    v2f b; b.x = p[0]; b.y = p[ldb]; return b;
}

// ---------------- K0: per-batch mean of pos (B x 4096 x 3) ----------------
__global__ void pos_mean_kernel(const float* __restrict__ pos, float* __restrict__ mean) {
    __shared__ float red[256 * 3];
    int t = threadIdx.x, b = blockIdx.x;
    float sx = 0.f, sy = 0.f, sz = 0.f;
    for (int i = t; i < NPTS; i += 256) {
        const float* p = pos + ((size_t)b * NPTS + i) * 3;
        sx += p[0]; sy += p[1]; sz += p[2];
    }
    red[t] = sx; red[256 + t] = sy; red[512 + t] = sz;
    __syncthreads();
    for (int s = 128; s > 0; s >>= 1) {
        if (t < s) {
            red[t]       += red[t + s];
            red[256 + t] += red[256 + t + s];
            red[512 + t] += red[512 + t + s];
        }
        __syncthreads();
    }
    if (t == 0) {
        mean[b * 3 + 0] = red[0]   * (1.0f / NPTS);
        mean[b * 3 + 1] = red[256] * (1.0f / NPTS);
        mean[b * 3 + 2] = red[512] * (1.0f / NPTS);
    }
}

// ---------------- K1: xp = x + rel @ W_pe + b_pe ----------------
__global__ void pe_add_kernel(const float* __restrict__ x, const float* __restrict__ pos,
                              const float* __restrict__ mean, const float* __restrict__ Wpe,
                              const float* __restrict__ bpe, float* __restrict__ xp) {
    size_t gid = (size_t)blockIdx.x * blockDim.x + threadIdx.x;
    if (gid >= SZMAT) return;
    int d    = (int)(gid & (DIMX - 1));
    size_t i = gid >> 9;                 // row (DIMX == 512)
    int b    = (int)(i >> 12);           // NPTS == 4096
    const float* p = pos + i * 3;
    float rx = p[0] - mean[b * 3 + 0];
    float ry = p[1] - mean[b * 3 + 1];
    float rz = p[2] - mean[b * 3 + 2];
    xp[gid] = x[gid] + rx * Wpe[d] + ry * Wpe[DIMX + d] + rz * Wpe[2 * DIMX + d] + bpe[d];
}

// ------- K2: qkv GEMM (8192x512)@(512x1536) + bias, de-interleave to q/k/v (B,H,N,E) -------
// One wave computes a 64x16 tile: B fragment loaded once per k-step, reused by 4 WMMAs.
__global__ void qkv_gemm_kernel(const float* __restrict__ xp, const float* __restrict__ Wqkv,
                                const float* __restrict__ bqkv,
                                float* __restrict__ qb, float* __restrict__ kb,
                                float* __restrict__ vb) {
    int lane = threadIdx.x;
    int col0 = blockIdx.x * 16;          // 0..1535
    int row0 = blockIdx.y * 64;          // 0..8191, 64-row strip
    v8f c0 = {}, c1 = {}, c2 = {}, c3 = {};
    for (int kk = 0; kk < DIMX; kk += 4) {
        v2f b  = frag_b(Wqkv, 3 * DIMX, col0, kk, lane);
        v2f a0 = frag_a(xp, DIMX, row0,      kk, lane);
        v2f a1 = frag_a(xp, DIMX, row0 + 16, kk, lane);
        v2f a2 = frag_a(xp, DIMX, row0 + 32, kk, lane);
        v2f a3 = frag_a(xp, DIMX, row0 + 48, kk, lane);
        c0 = __builtin_amdgcn_wmma_f32_16x16x4_f32(false, a0, false, b, (short)0, c0, false, false);
        c1 = __builtin_amdgcn_wmma_f32_16x16x4_f32(false, a1, false, b, (short)0, c1, false, false);
        c2 = __builtin_amdgcn_wmma_f32_16x16x4_f32(false, a2, false, b, (short)0, c2, false, false);
        c3 = __builtin_amdgcn_wmma_f32_16x16x4_f32(false, a3, false, b, (short)0, c3, false, false);
    }
    // column -> (h, e, which) with K innermost: col = h*192 + e*3 + which
    int nn = lane & 15, kh = lane >> 4;
    int col = col0 + nn;
    int h   = col / (HE * 3);
    int rem = col - h * (HE * 3);
    int e   = rem / 3;
    int w   = rem - e * 3;
    float bias = bqkv[col];
    float* dst = (w == 0) ? qb : (w == 1) ? kb : vb;
    int bidx = row0 >> 12;               // batch (64-row strip never crosses batch)
    int nq0  = row0 & (NPTS - 1);
    size_t base = ((size_t)(bidx * NH + h) * NPTS + nq0) * HE + e;
    v8f acc[4] = {c0, c1, c2, c3};
    for (int t = 0; t < 4; ++t)
        for (int r = 0; r < 8; ++r) {
            int rr = t * 16 + r + 8 * kh;
            dst[base + (size_t)rr * HE] = acc[t][r] + bias;
        }
}

// ------- K3: sel_keys GEMM (4096x1024)@(1024x64) + b_sel -> selk (B*H*n, 64) -------
// One wave computes a 64x16 tile of the (4096 x 64) result.
__global__ void sel_gemm_kernel(const float* __restrict__ kbuf, const float* __restrict__ Wsel,
                                const float* __restrict__ bsel, float* __restrict__ selk) {
    int lane = threadIdx.x;
    int col0 = blockIdx.x * 16;          // 0..63
    int row0 = blockIdx.y * 64;          // 0..4095 (64 | 256 -> strip stays in one bh)
    int bh   = row0 >> 8;
    const float* A = kbuf + (size_t)bh * NPTS * HE;   // ball rows are 1024 contiguous floats
    int r0 = row0 & 255;
    v8f c0 = {}, c1 = {}, c2 = {}, c3 = {};
    for (int kk = 0; kk < MSEL * HE; kk += 4) {
        v2f b  = frag_b(Wsel, HE, col0, kk, lane);
        v2f a0 = frag_a(A, MSEL * HE, r0,      kk, lane);
        v2f a1 = frag_a(A, MSEL * HE, r0 + 16, kk, lane);
        v2f a2 = frag_a(A, MSEL * HE, r0 + 32, kk, lane);
        v2f a3 = frag_a(A, MSEL * HE, r0 + 48, kk, lane);
        c0 = __builtin_amdgcn_wmma_f32_16x16x4_f32(false, a0, false, b, (short)0, c0, false, false);
        c1 = __builtin_amdgcn_wmma_f32_16x16x4_f32(false, a1, false, b, (short)0, c1, false, false);
        c2 = __builtin_amdgcn_wmma_f32_16x16x4_f32(false, a2, false, b, (short)0, c2, false, false);
        c3 = __builtin_amdgcn_wmma_f32_16x16x4_f32(false, a3, false, b, (short)0, c3, false, false);
    }
    int nn = lane & 15, kh = lane >> 4;
    float bias = bsel[col0 + nn];
    v8f acc[4] = {c0, c1, c2, c3};
    for (int t = 0; t < 4; ++t)
        for (int r = 0; r < 8; ++r) {
            int row = row0 + t * 16 + r + 8 * kh;
            selk[(size_t)row * HE + col0 + nn] = acc[t][r] + bias;
        }
}

// ------- K4: sim = q @ selk^T (WMMA), block-mask, top-2 per query -------
// Q tile (16x64) preloaded into 16 A-fragments, reused across all 16 ball-column tiles.
__global__ void sim_top2_kernel(const float* __restrict__ qbuf, const float* __restrict__ selk,
                                int* __restrict__ idx) {
    __shared__ float sim[16 * NBALL];    // 16 KB
    int lane = threadIdx.x;
    int blk  = blockIdx.x;               // bh*256 + qtile
    int bh   = blk >> 8;
    int qt   = blk & 255;
    const float* Q = qbuf + (size_t)bh * NPTS * HE;
    const float* S = selk + (size_t)bh * NBALL * HE;
    int m = lane & 15, kh = lane >> 4;
    v2f aq[16];
    #pragma unroll
    for (int ks = 0; ks < 16; ++ks) {
        const float* pa = Q + (size_t)(qt * 16 + m) * HE + ks * 4 + 2 * kh;
        aq[ks].x = pa[0]; aq[ks].y = pa[1];
    }
    for (int ct = 0; ct < NBALL / 16; ++ct) {
        v8f c = {};
        #pragma unroll
        for (int ks = 0; ks < 16; ++ks) {
            // B[k][n] = selk[n][k]  (selk rows are the balls)
            const float* pb = S + (size_t)(ct * 16 + m) * HE + ks * 4 + 2 * kh;
            v2f b; b.x = pb[0]; b.y = pb[1];
            c = __builtin_amdgcn_wmma_f32_16x16x4_f32(false, aq[ks], false, b, (short)0, c, false, false);
        }
        for (int r = 0; r < 8; ++r)
            sim[(r + 8 * kh) * NBALL + ct * 16 + m] = c[r];
    }
    __syncthreads();
    if (lane < 16) {
        int qg = qt * 16 + lane;
        int mb = (qg >> 6) << 2;         // own 64-point block == 4 consecutive balls
        float b1 = -__builtin_inff(), b2 = -__builtin_inff();
        int i1 = 0, i2 = 0;
        const float* row = &sim[lane * NBALL];
        for (int i = 0; i < NBALL; ++i) {
            if (i >= mb && i < mb + 4) continue;
            float s = row[i];
            if (s > b1)      { b2 = b1; i2 = i1; b1 = s; i1 = i; }
            else if (s > b2) { b2 = s; i2 = i; }
        }
        idx[((size_t)bh * NPTS + qg) * 2 + 0] = i1;
        idx[((size_t)bh * NPTS + qg) * 2 + 1] = i2;
    }
}

// ------- K5: gather 2 balls, 32-wide softmax attention per query (one wave/query) -------
__global__ void attn_kernel(const float* __restrict__ qbuf, const float* __restrict__ kbuf,
                            const float* __restrict__ vbuf, const int* __restrict__ idx,
                            float* __restrict__ aout) {
    __shared__ float wsh[32];
    __shared__ int   rowk[32];
    int lane = threadIdx.x;
    int gq   = blockIdx.x;               // bh*4096 + qn
    int bh   = gq >> 12;
    int qn   = gq & (NPTS - 1);
    const float4* qv = (const float4*)(qbuf + ((size_t)bh * NPTS + qn) * HE);
    int ball = idx[(size_t)gq * 2 + (lane >> 4)];
    int krow = bh * NPTS + ball * MSEL + (lane & 15);
    const float4* kv = (const float4*)(kbuf + (size_t)krow * HE);
    float s = 0.f;
    for (int t = 0; t < HE / 4; ++t) {
        float4 a = qv[t], b = kv[t];
        s += a.x * b.x + a.y * b.y + a.z * b.z + a.w * b.w;
    }
    s *= 0.125f;                          // 1/sqrt(64)
    float mx = s;
    for (int off = 16; off > 0; off >>= 1) mx = fmaxf(mx, __shfl_xor(mx, off, 32));
    float p = __expf(s - mx);
    float sum = p;
    for (int off = 16; off > 0; off >>= 1) sum += __shfl_xor(sum, off, 32);
    wsh[lane]  = p / sum;
    rowk[lane] = krow;
    __syncthreads();
    float acc0 = 0.f, acc1 = 0.f;
    int e = lane;
    for (int j = 0; j < 32; ++j) {
        float wj = wsh[j];
        const float* vr = vbuf + (size_t)rowk[j] * HE;
        acc0 += wj * vr[e];
        acc1 += wj * vr[e + 32];
    }
    int b = bh >> 3, h = bh & 7;
    size_t o = (((size_t)b * NPTS + qn) * NH + h) * HE;   // (b, n, h, e) layout
    aout[o + e]      = acc0;
    aout[o + e + 32] = acc1;
}

// ------- K6: out = aout @ W_proj + b_proj  (8192x512)@(512x512), 64x16 tile per wave -------
__global__ void proj_gemm_kernel(const float* __restrict__ aout, const float* __restrict__ Wp,
                                 const float* __restrict__ bp, float* __restrict__ out) {
    int lane = threadIdx.x;
    int col0 = blockIdx.x * 16;
    int row0 = blockIdx.y * 64;
    v8f c0 = {}, c1 = {}, c2 = {}, c3 = {};
    for (int kk = 0; kk < DIMX; kk += 4) {
        v2f b  = frag_b(Wp, DIMX, col0, kk, lane);
        v2f a0 = frag_a(aout, DIMX, row0,      kk, lane);
        v2f a1 = frag_a(aout, DIMX, row0 + 16, kk, lane);
        v2f a2 = frag_a(aout, DIMX, row0 + 32, kk, lane);
        v2f a3 = frag_a(aout, DIMX, row0 + 48, kk, lane);
        c0 = __builtin_amdgcn_wmma_f32_16x16x4_f32(false, a0, false, b, (short)0, c0, false, false);
        c1 = __builtin_amdgcn_wmma_f32_16x16x4_f32(false, a1, false, b, (short)0, c1, false, false);
        c2 = __builtin_amdgcn_wmma_f32_16x16x4_f32(false, a2, false, b, (short)0, c2, false, false);
        c3 = __builtin_amdgcn_wmma_f32_16x16x4_f32(false, a3, false, b, (short)0, c3, false, false);
    }
    int nn = lane & 15, kh = lane >> 4;
    float bias = bp[col0 + nn];
    v8f acc[4] = {c0, c1, c2, c3};
    for (int t = 0; t < 4; ++t)
        for (int r = 0; r < 8; ++r) {
            int row = row0 + t * 16 + r + 8 * kh;
            out[(size_t)row * DIMX + col0 + nn] = acc[t][r] + bias;
        }
}

extern "C" void kernel_launch(void* const* d_in, const int* in_sizes, int n_in,
                              void* d_out, int out_size, void* d_ws, size_t ws_size,
                              hipStream_t stream) {
    const float* x     = (const float*)d_in[0];
    const float* pos   = (const float*)d_in[1];
    const float* W_pe  = (const float*)d_in[2];
    const float* b_pe  = (const float*)d_in[3];
    const float* W_qkv = (const float*)d_in[4];
    const float* b_qkv = (const float*)d_in[5];
    const float* W_sel = (const float*)d_in[6];
    const float* b_sel = (const float*)d_in[7];
    const float* W_prj = (const float*)d_in[8];
    const float* b_prj = (const float*)d_in[9];
    float* out = (float*)d_out;

    // workspace layout (floats): ~87 MB total
    float* ws   = (float*)d_ws;
    float* mean = ws;                    // 16 (padded)
    float* xp   = ws + 16;               // SZMAT
    float* qb   = xp + SZMAT;            // SZMAT
    float* kb   = qb + SZMAT;            // SZMAT
    float* vb   = kb + SZMAT;            // SZMAT
    float* selk = vb + SZMAT;            // B*H*n*E = 262144
    float* aout = selk + (size_t)NB * NH * NBALL * HE;  // SZMAT
    int*   idx  = (int*)(aout + SZMAT);  // B*H*N*2 = 131072 ints

    pos_mean_kernel<<<NB, 256, 0, stream>>>(pos, mean);
    pe_add_kernel<<<(int)(SZMAT / 256), 256, 0, stream>>>(x, pos, mean, W_pe, b_pe, xp);
    qkv_gemm_kernel<<<dim3(3 * DIMX / 16, SZROW / 64), 32, 0, stream>>>(xp, W_qkv, b_qkv, qb, kb, vb);
    sel_gemm_kernel<<<dim3(HE / 16, NB * NH * NBALL / 64), 32, 0, stream>>>(kb, W_sel, b_sel, selk);
    sim_top2_kernel<<<NB * NH * (NPTS / 16), 32, 0, stream>>>(qb, selk, idx);
    attn_kernel<<<NB * NH * NPTS, 32, 0, stream>>>(qb, kb, vb, idx, aout);
    proj_gemm_kernel<<<dim3(DIMX / 16, SZROW / 64), 32, 0, stream>>>(aout, W_prj, b_prj, out);
}